// DepthwiseConv1Version_9912784520076
// MI455X (gfx1250) — compile-verified
//
#include <hip/hip_runtime.h>
#include <hip/hip_bf16.h>
#include <stdint.h>

// ---------------------------------------------------------------------------
// Depthwise 3x3 conv (C=64, N=32, H=W=128) + bias + BatchNorm(train) + ReLU.
// Memory-bound (~3 FLOP/byte @ 23.3 TB/s => ~11.5us floor): optimize data
// movement, not math.
//  Pass 1: TDM tensor_load_to_lds stages each 128x128 plane (8 waves x
//          16-row slabs) into LDS; each thread computes 4 adjacent pixels of
//          the zero-padded 3x3 stencil from LDS (b128 DS loads, b128 stores),
//          accumulates per-channel sum / sum^2 (block reduce + 2 atomics).
//  Pass 2: fold stats -> per-channel scale/shift.
//  Pass 3: in-place float4 normalize+ReLU on d_out (L2-resident reread),
//          with global_prefetch one stride ahead.
// ---------------------------------------------------------------------------

#define N_     32
#define C_     64
#define H_     128
#define W_     128
#define PLANE  (H_ * W_)            // 16384
#define NHW    (N_ * PLANE)         // 524288
#define INV_NHW (1.0f / (float)NHW)
#define EPSbn  1e-5f

typedef __attribute__((ext_vector_type(4))) unsigned int v4u;
typedef __attribute__((ext_vector_type(4))) int          v4i;
typedef __attribute__((ext_vector_type(8))) int          v8i;

// Issue one TDM 2D tile load: `rows` rows of 128 fp32 from global -> LDS.
// D# per CDNA5 ISA ch.8: group0 = {count, lds_addr, global_addr, type=2},
// group1 = {data_size=4B, tensor_dim0=128, tensor_dim1=rows, tile 128 x rows,
// tensor_dim0_stride=128}. Groups 2/3 zero (2D tensor).
__device__ __forceinline__ void tdm_load_rows(const float* gptr,
                                              unsigned lds_byte_addr,
                                              unsigned rows) {
  unsigned long long ga = (unsigned long long)(uintptr_t)gptr;
  v4u g0;
  g0.x = 1u;                                           // count=1, user mode
  g0.y = lds_byte_addr;                                // lds_addr
  g0.z = (unsigned)(ga & 0xFFFFFFFFull);               // global_addr[31:0]
  g0.w = (unsigned)((ga >> 32) & 0x1FFFFFFull)         // global_addr[56:32]
         | (2u << 30);                                 // type = 2 ("image")
  v8i g1;
  g1.s0 = (int)(2u << 16);           // workgroup_mask=0, data_size=2 (4 bytes)
  g1.s1 = (int)(128u << 16);         // tensor_dim0[15:0] = 128 (bits 63:48)
  g1.s2 = (int)(rows << 16);         // tensor_dim1[15:0] = rows (bits 95:80)
  g1.s3 = (int)(128u << 16);         // tile_dim0 = 128 (bits 127:112)
  g1.s4 = (int)rows;                 // tile_dim1 = rows, tile_dim2 = 0
  g1.s5 = 128;                       // tensor_dim0_stride[31:0] = 128 elems
  g1.s6 = 0;                         // dim0_stride[47:32]=0, dim1_stride lo=0
  g1.s7 = 0;
  v4i z4 = {0, 0, 0, 0};
#if defined(__clang_major__) && (__clang_major__ >= 23)
  v8i z8 = {0, 0, 0, 0, 0, 0, 0, 0};
  __builtin_amdgcn_tensor_load_to_lds(g0, g1, z4, z4, z8, 0);
#else
  __builtin_amdgcn_tensor_load_to_lds(g0, g1, z4, z4, 0);
#endif
}

__global__ void dwbn_init_kernel(float* ws) {
  int t = threadIdx.x;
  if (t < 2 * C_) ws[t] = 0.0f;     // ws[0:64)=sum, ws[64:128)=sumsq
}

// One block per (n,c) plane. 256 threads = 8 waves; each wave TDM-loads a
// 16-row slab of the plane into LDS; each thread then computes 4 adjacent
// pixels per iteration (b128 LDS loads + b128 stores), and the block reduces
// sum / sum^2 into global accumulators.
__global__ void __launch_bounds__(256)
dwbn_conv_stats_kernel(const float* __restrict__ x,
                       const float* __restrict__ w,
                       const float* __restrict__ b,
                       float* __restrict__ y,
                       float* __restrict__ ws) {
  __shared__ __align__(16) float tile[PLANE];  // 64 KB plane (reused f. reduce)

  const int plane = blockIdx.x;     // 0 .. N*C-1
  const int c     = plane & (C_ - 1);
  const int t     = threadIdx.x;
  const float* xp = x + (size_t)plane * PLANE;
  float*       yp = y + (size_t)plane * PLANE;

  // --- async TDM stage: each wave pulls its own 16-row slab ---------------
  {
    unsigned wave = (unsigned)__builtin_amdgcn_readfirstlane((int)(t >> 5));
    unsigned lds_base = (unsigned)(unsigned long long)(uintptr_t)&tile[0];
    const float* gsrc = xp + wave * 16u * W_;
    tdm_load_rows(gsrc, lds_base + wave * 16u * W_ * 4u, 16u);
    __builtin_amdgcn_s_wait_tensorcnt(0);   // this wave's slab landed
  }
  __syncthreads();                          // whole plane resident in LDS

  // per-channel 3x3 weights + bias: wave-uniform -> scalar loads
  const float* wc = w + c * 9;
  const float w00 = wc[0], w01 = wc[1], w02 = wc[2];
  const float w10 = wc[3], w11 = wc[4], w12 = wc[5];
  const float w20 = wc[6], w21 = wc[7], w22 = wc[8];
  const float bias = b[c];

  const float4 zero4 = {0.f, 0.f, 0.f, 0.f};
  float lsum = 0.0f, lsq = 0.0f;

  // 16384 pixels / (256 threads * 4 px) = 16 iterations
#pragma unroll 4
  for (int k = 0; k < PLANE / (256 * 4); ++k) {
    const int base = (k << 10) + (t << 2);  // 4 adjacent pixels, 16B aligned
    const int r    = base >> 7;
    const int c0   = base & (W_ - 1);       // multiple of 4

    const bool rm = (r > 0), rp = (r < H_ - 1);
    const bool cm = (c0 > 0), cp = (c0 < W_ - 4);

    // row segments [c0-1 .. c0+4]: one b128 + two guarded b32 per row
    const float* r0 = &tile[(r - 1) * W_ + c0];
    const float* r1 = &tile[r * W_ + c0];
    const float* r2 = &tile[(r + 1) * W_ + c0];

    float4 tM = rm ? *(const float4*)r0 : zero4;
    float  tL = (rm && cm) ? r0[-1] : 0.f;
    float  tR = (rm && cp) ? r0[4]  : 0.f;
    float4 mM = *(const float4*)r1;
    float  mL = cm ? r1[-1] : 0.f;
    float  mR = cp ? r1[4]  : 0.f;
    float4 bM = rp ? *(const float4*)r2 : zero4;
    float  bL = (rp && cm) ? r2[-1] : 0.f;
    float  bR = (rp && cp) ? r2[4]  : 0.f;

    float4 o;
    o.x = bias;
    o.x = fmaf(w00, tL,   o.x); o.x = fmaf(w01, tM.x, o.x); o.x = fmaf(w02, tM.y, o.x);
    o.x = fmaf(w10, mL,   o.x); o.x = fmaf(w11, mM.x, o.x); o.x = fmaf(w12, mM.y, o.x);
    o.x = fmaf(w20, bL,   o.x); o.x = fmaf(w21, bM.x, o.x); o.x = fmaf(w22, bM.y, o.x);
    o.y = bias;
    o.y = fmaf(w00, tM.x, o.y); o.y = fmaf(w01, tM.y, o.y); o.y = fmaf(w02, tM.z, o.y);
    o.y = fmaf(w10, mM.x, o.y); o.y = fmaf(w11, mM.y, o.y); o.y = fmaf(w12, mM.z, o.y);
    o.y = fmaf(w20, bM.x, o.y); o.y = fmaf(w21, bM.y, o.y); o.y = fmaf(w22, bM.z, o.y);
    o.z = bias;
    o.z = fmaf(w00, tM.y, o.z); o.z = fmaf(w01, tM.z, o.z); o.z = fmaf(w02, tM.w, o.z);
    o.z = fmaf(w10, mM.y, o.z); o.z = fmaf(w11, mM.z, o.z); o.z = fmaf(w12, mM.w, o.z);
    o.z = fmaf(w20, bM.y, o.z); o.z = fmaf(w21, bM.z, o.z); o.z = fmaf(w22, bM.w, o.z);
    o.w = bias;
    o.w = fmaf(w00, tM.z, o.w); o.w = fmaf(w01, tM.w, o.w); o.w = fmaf(w02, tR,   o.w);
    o.w = fmaf(w10, mM.z, o.w); o.w = fmaf(w11, mM.w, o.w); o.w = fmaf(w12, mR,   o.w);
    o.w = fmaf(w20, bM.z, o.w); o.w = fmaf(w21, bM.w, o.w); o.w = fmaf(w22, bR,   o.w);

    *(float4*)&yp[base] = o;               // coalesced b128 store

    lsum += (o.x + o.y) + (o.z + o.w);
    lsq   = fmaf(o.x, o.x, lsq);
    lsq   = fmaf(o.y, o.y, lsq);
    lsq   = fmaf(o.z, o.z, lsq);
    lsq   = fmaf(o.w, o.w, lsq);
  }

  // --- block reduction (reuse tile after all LDS stencil reads done) ------
  __syncthreads();
  tile[t]       = lsum;
  tile[256 + t] = lsq;
  __syncthreads();
#pragma unroll
  for (int s = 128; s > 0; s >>= 1) {
    if (t < s) {
      tile[t]       += tile[t + s];
      tile[256 + t] += tile[256 + t + s];
    }
    __syncthreads();
  }
  if (t == 0) {
    atomicAdd(&ws[c],      tile[0]);
    atomicAdd(&ws[C_ + c], tile[256]);
  }
}

__global__ void dwbn_stats_kernel(const float* __restrict__ ws_in,
                                  float* __restrict__ scale,
                                  float* __restrict__ shift,
                                  const float* __restrict__ gamma,
                                  const float* __restrict__ beta) {
  int c = threadIdx.x;
  if (c < C_) {
    float mean = ws_in[c] * INV_NHW;
    float var  = fmaf(-mean, mean, ws_in[C_ + c] * INV_NHW);  // biased var
    float s    = gamma[c] * rsqrtf(var + EPSbn);
    scale[c]   = s;
    shift[c]   = fmaf(-mean, s, beta[c]);
  }
}

__global__ void __launch_bounds__(256)
dwbn_bnrelu_kernel(float* __restrict__ y,
                   const float* __restrict__ scale,
                   const float* __restrict__ shift,
                   int total4) {
  int i      = blockIdx.x * blockDim.x + threadIdx.x;
  int stride = gridDim.x * blockDim.x;
  float4* p  = (float4*)y;
  for (; i < total4; i += stride) {
    __builtin_prefetch(&p[i + stride], 0, 0);  // global_prefetch_b8 ahead
    int c    = (i >> 12) & (C_ - 1);    // 4 consecutive floats share channel
    float s  = scale[c];
    float sh = shift[c];
    float4 v = p[i];
    v.x = fmaxf(fmaf(v.x, s, sh), 0.0f);
    v.y = fmaxf(fmaf(v.y, s, sh), 0.0f);
    v.z = fmaxf(fmaf(v.z, s, sh), 0.0f);
    v.w = fmaxf(fmaf(v.w, s, sh), 0.0f);
    p[i] = v;
  }
}

extern "C" void kernel_launch(void* const* d_in, const int* in_sizes, int n_in,
                              void* d_out, int out_size, void* d_ws, size_t ws_size,
                              hipStream_t stream) {
  const float* x     = (const float*)d_in[0];
  const float* w     = (const float*)d_in[1];
  const float* b     = (const float*)d_in[2];
  const float* gamma = (const float*)d_in[3];
  const float* beta  = (const float*)d_in[4];
  float* out = (float*)d_out;
  float* ws  = (float*)d_ws;   // [0:64) sum, [64:128) sumsq, [128:192) scale, [192:256) shift

  dwbn_init_kernel<<<1, 128, 0, stream>>>(ws);

  dwbn_conv_stats_kernel<<<N_ * C_, 256, 0, stream>>>(x, w, b, out, ws);

  dwbn_stats_kernel<<<1, 64, 0, stream>>>(ws, ws + 2 * C_, ws + 3 * C_,
                                          gamma, beta);

  int total4 = out_size / 4;   // 8,388,608 float4s
  dwbn_bnrelu_kernel<<<8192, 256, 0, stream>>>(out, ws + 2 * C_, ws + 3 * C_,
                                               total4);
}